// KnowledgeGraphGNN_9663676416485
// MI455X (gfx1250) — compile-verified
//
#include <hip/hip_runtime.h>

// ---------------------------------------------------------------------------
// KnowledgeGraphGNN, algebraically collapsed for MI455X (gfx1250).
//
//   out[b,:] = (sum_j c[j] * relu(M1b[j,:] + 0.1*u[b,j]*s1[:])) @ W2 + b2
//   u[b]     = Ahat @ softmax(latent[b,0:8])
//   M1b      = Ahat @ (NF @ W1) + b1      (constant, 8x256)
//   s1       = colsum(W1), c = (1/8) colsum(Ahat)
//
// FLOPs after collapse: ~0.54 GFLOP (one [8192x256][256x128] fp32 GEMM).
// Bytes: ~4.5 MB  ->  bandwidth-trivial at 23.3 TB/s; W2 (128 KB) lives in L2.
// GEMM runs on the matrix pipe via V_WMMA_F32_16X16X4_F32 (fp32, matches the
// fp32 reference; RNE rounding per ISA 7.12).
// ---------------------------------------------------------------------------

typedef __attribute__((ext_vector_type(2))) float v2f;
typedef __attribute__((ext_vector_type(8))) float v8f;

#define NND   8      // nodes
#define FDIM  512    // input features
#define HDIM  256    // hidden
#define ODIM  128    // output
#define BROWS 16     // batch rows per workgroup (WMMA M)
#define HT_STRIDE 257  // odd stride -> conflict-free LDS column reads

// workspace layout (floats)
#define WS_AHAT 0      //  64: Ahat[8][8]
#define WS_C    64     //   8: c[8]
#define WS_S1   128    // 256: s1
#define WS_M1B  512    // 2048: M1b[8][256]
// total: 2560 floats = 10 KB

// ---------------------------------------------------------------------------
// Prep: Ahat, c, s1, M1b  (one block, 256 threads; ~1M FMA total)
// ---------------------------------------------------------------------------
__global__ __launch_bounds__(HDIM) void gnn_prep(
    const float* __restrict__ nf,   // [8,512]
    const float* __restrict__ ew,   // [56]
    const float* __restrict__ W1,   // [512,256]
    const float* __restrict__ b1,   // [256]
    float* __restrict__ ws)
{
    __shared__ float sA[NND * NND];
    const int tid = threadIdx.x;

    if (tid == 0) {
        float A[NND][NND];
        for (int i = 0; i < NND; ++i)
            for (int j = 0; j < NND; ++j) A[i][j] = (i == j) ? 1.0f : 0.0f;
        int e = 0;
        // edges (i,j), i!=j, row-major; A[DST,SRC] = w  ->  A[j][i]
        for (int i = 0; i < NND; ++i)
            for (int j = 0; j < NND; ++j)
                if (i != j) { A[j][i] += ew[e]; ++e; }
        float dinv[NND];
        for (int r = 0; r < NND; ++r) {
            float d = 0.f;
            for (int k = 0; k < NND; ++k) d += A[r][k];
            dinv[r] = (d > 0.f) ? (1.0f / sqrtf(d)) : 0.0f;
        }
        for (int i = 0; i < NND; ++i)
            for (int j = 0; j < NND; ++j)
                sA[i * NND + j] = dinv[i] * A[i][j] * dinv[j];
    }
    __syncthreads();

    // each thread owns one hidden column h = tid
    const int h = tid;
    float s1 = 0.f;
    float T[NND];
    #pragma unroll
    for (int j = 0; j < NND; ++j) T[j] = 0.f;
    for (int f = 0; f < FDIM; ++f) {
        const float wv = W1[f * HDIM + h];      // coalesced across threads
        s1 += wv;
        #pragma unroll
        for (int j = 0; j < NND; ++j)           // nf[..] uniform -> scalar loads
            T[j] = fmaf(nf[j * FDIM + f], wv, T[j]);
    }
    ws[WS_S1 + h] = s1;

    const float bb = b1[h];
    #pragma unroll
    for (int j = 0; j < NND; ++j) {
        float m = bb;
        #pragma unroll
        for (int k = 0; k < NND; ++k) m = fmaf(sA[j * NND + k], T[k], m);
        ws[WS_M1B + j * HDIM + h] = m;
    }

    if (tid < NND * NND) ws[WS_AHAT + tid] = sA[tid];
    if (tid < NND) {
        float cc = 0.f;
        for (int i = 0; i < NND; ++i) cc += sA[i * NND + tid];
        ws[WS_C + tid] = 0.125f * cc;
    }
}

// ---------------------------------------------------------------------------
// Main: 16 batch rows per block, 8 waves; wave w owns output cols [16w,16w+16)
// ---------------------------------------------------------------------------
__global__ __launch_bounds__(256) void gnn_main(
    const float* __restrict__ latent,  // [B,512]
    const float* __restrict__ W2,      // [256,128]
    const float* __restrict__ b2,      // [128]
    const float* __restrict__ ws,
    float* __restrict__ out,           // [B,128]
    int B)
{
    __shared__ float sAhat[NND * NND];
    __shared__ float sC[NND];
    __shared__ float sS1[HDIM];
    __shared__ float sM1b[NND * HDIM];
    __shared__ float sU[BROWS][NND];
    __shared__ float sHt[BROWS][HT_STRIDE];

    const int tid = threadIdx.x;
    const int b0  = blockIdx.x * BROWS;

    // stage constants into LDS (tiny, L2-hot)
    if (tid < NND * NND) sAhat[tid] = ws[WS_AHAT + tid];
    if (tid < NND)       sC[tid]    = ws[WS_C + tid];
    sS1[tid] = ws[WS_S1 + tid];
    #pragma unroll
    for (int r = 0; r < NND; ++r)
        sM1b[r * HDIM + tid] = ws[WS_M1B + r * HDIM + tid];
    __syncthreads();

    // phase 1a: softmax over latent[b,0:8] and u = Ahat @ w (one row/thread)
    if (tid < BROWS) {
        int row = b0 + tid; if (row >= B) row = B - 1;
        const float* lp = latent + (size_t)row * FDIM;
        float x[NND];
        float mx = -INFINITY;
        #pragma unroll
        for (int j = 0; j < NND; ++j) { x[j] = lp[j]; mx = fmaxf(mx, x[j]); }
        float s = 0.f;
        #pragma unroll
        for (int j = 0; j < NND; ++j) { x[j] = __expf(x[j] - mx); s += x[j]; }
        const float inv = 0.1f / s;              // fold 0.1 and softmax denom
        #pragma unroll
        for (int i = 0; i < NND; ++i) {
            float u = 0.f;
            #pragma unroll
            for (int j = 0; j < NND; ++j) u = fmaf(sAhat[i * NND + j], x[j], u);
            sU[tid][i] = u * inv;
        }
    }
    __syncthreads();

    // phase 1b: Ht[r][h] = sum_j c[j]*relu(M1b[j,h] + u[r,j]*s1[h]); h = tid
    {
        const float s1v = sS1[tid];
        for (int r = 0; r < BROWS; ++r) {
            float acc = 0.f;
            #pragma unroll
            for (int j = 0; j < NND; ++j) {
                const float v = fmaf(sU[r][j], s1v, sM1b[j * HDIM + tid]);
                acc = fmaf(sC[j], fmaxf(v, 0.f), acc);
            }
            sHt[r][tid] = acc;
        }
    }
    __syncthreads();

    // phase 2: D[16x16] += A[16x4] x B[4x16] over K=256 (fp32 WMMA)
    const int wave  = tid >> 5;
    const int lane  = tid & 31;
    const int n0    = wave * 16;          // output-column tile
    const int lh    = lane & 15;
    const bool hi   = lane >= 16;

    v8f acc = {};
    #pragma unroll 8
    for (int k0 = 0; k0 < HDIM; k0 += 4) {
        const int ka = k0 + (hi ? 2 : 0);
        v2f a, b;
        // A 16x4 fp32 layout: lanes 0-15 M=lane {K0,K1}; lanes 16-31 M=lane-16 {K2,K3}
        a.x = sHt[lh][ka];
        a.y = sHt[lh][ka + 1];
        // B 4x16 fp32 (mirror): lanes 0-15 N=lane {K0,K1}; lanes 16-31 N=lane-16 {K2,K3}
        b.x = W2[(size_t)ka * ODIM + n0 + lh];
        b.y = W2[(size_t)(ka + 1) * ODIM + n0 + lh];
        acc = __builtin_amdgcn_wmma_f32_16x16x4_f32(
            /*neg_a=*/false, a, /*neg_b=*/false, b,
            /*c_mod=*/(short)0, acc, /*reuse_a=*/false, /*reuse_b=*/false);
    }

    // D 16x16 f32 layout: VGPR v -> M = v + (hi?8:0), N = lane (mod 16)
    const float bias = b2[n0 + lh];
    float* op = out + (size_t)(b0 + (hi ? 8 : 0)) * ODIM + n0 + lh;

    if (b0 + BROWS <= B) {
        // full tile (always taken for B % 16 == 0): straight-line stores,
        // uniform branch -> EXEC stays all-ones, no per-element cmp/branch
        #pragma unroll
        for (int v = 0; v < 8; ++v)
            op[(size_t)v * ODIM] = acc[v] + bias;
    } else {
        #pragma unroll
        for (int v = 0; v < 8; ++v) {
            const int row = b0 + (hi ? 8 : 0) + v;
            if (row < B)
                out[(size_t)row * ODIM + n0 + lh] = acc[v] + bias;
        }
    }
}

// ---------------------------------------------------------------------------
extern "C" void kernel_launch(void* const* d_in, const int* in_sizes, int n_in,
                              void* d_out, int out_size, void* d_ws, size_t ws_size,
                              hipStream_t stream)
{
    const float* latent = (const float*)d_in[0];  // [B,512]
    const float* nf     = (const float*)d_in[1];  // [8,512]
    const float* ew     = (const float*)d_in[2];  // [56,1]
    const float* W1     = (const float*)d_in[3];  // [512,256]
    const float* b1     = (const float*)d_in[4];  // [256]
    const float* W2     = (const float*)d_in[5];  // [256,128]
    const float* b2     = (const float*)d_in[6];  // [128]
    float* out = (float*)d_out;
    float* ws  = (float*)d_ws;                    // needs 10 KB

    const int B = in_sizes[0] / FDIM;             // 8192

    gnn_prep<<<1, HDIM, 0, stream>>>(nf, ew, W1, b1, ws);
    gnn_main<<<(B + BROWS - 1) / BROWS, 256, 0, stream>>>(latent, W2, b2, ws, out, B);
}